// GNN_49744311222870
// MI455X (gfx1250) — compile-verified
//
#include <hip/hip_runtime.h>
#include <math.h>

// ---------------------------------------------------------------------------
// GNN (3-layer GCN) for MI455X / gfx1250.
// GEMMs use V_WMMA_F32_16X16X4_F32 (exact fp32, matches reference numerics).
// Edge aggregation = L2-resident fp32 atomic scatter (agg buffer 25.6MB << 192MB L2).
// B-matrix staged in LDS K-pair-interleaved so each WMMA B fragment is one
// conflict-free ds_load_b64 into an adjacent VGPR pair (no repack movs).
// ---------------------------------------------------------------------------

typedef __attribute__((ext_vector_type(2))) float v2f;
typedef __attribute__((ext_vector_type(8))) float v8f;

#define N_NODES 100000
#define N_EDGES 1600000
#define HID 64
// LDS row stride per K-pair: 64 cols * 2 floats + 32 pad words.
// 160 % 64 == 32  => lanes 0-15 (kh=0) and 16-31 (kh=1) hit disjoint bank halves.
#define PAIR_STRIDE 160

// ---------------------------- small helpers --------------------------------

__global__ void zero_f32_kernel(float* __restrict__ p, int n) {
  int i = blockIdx.x * blockDim.x + threadIdx.x;
  if (i < n) p[i] = 0.0f;
}

// deg[col[e]] += 1 over all edges (cols are int64 in the reference)
__global__ void deg_kernel(const long long* __restrict__ col, float* __restrict__ deg, int E) {
  int stride = gridDim.x * blockDim.x;
  for (int e = blockIdx.x * blockDim.x + threadIdx.x; e < E; e += stride) {
    atomicAdd(&deg[(int)col[e]], 1.0f);
  }
}

// dinv[i] = rsqrt(deg[i] + 1)   (+1 self loop; deg >= 0)
__global__ void dinv_kernel(float* __restrict__ d, int n) {
  int i = blockIdx.x * blockDim.x + threadIdx.x;
  if (i < n) d[i] = rsqrtf(d[i] + 1.0f);
}

// ---------------------------- edge scatter ---------------------------------
// One wave32 per edge: lane j handles columns 2j, 2j+1.
// agg[col] += t[row] * dinv[row]*dinv[col]
__global__ void scatter_kernel(const long long* __restrict__ rows,
                               const long long* __restrict__ cols,
                               const float* __restrict__ dinv,
                               const float* __restrict__ t,
                               float* __restrict__ agg, int E) {
  int lane = threadIdx.x & 31;
  int wave = (blockIdx.x * blockDim.x + threadIdx.x) >> 5;
  int nwaves = (gridDim.x * blockDim.x) >> 5;
  for (int e = wave; e < E; e += nwaves) {
    // prefetch next iteration's edge indices (global_prefetch_b8)
    if (e + nwaves < E) {
      __builtin_prefetch(&rows[e + nwaves], 0, 0);
      __builtin_prefetch(&cols[e + nwaves], 0, 0);
    }
    int r = (int)rows[e];
    int c = (int)cols[e];
    float coef = dinv[r] * dinv[c];
    const float2* trow = (const float2*)(t + (size_t)r * HID);
    float2 v = trow[lane];
    float* dst = agg + (size_t)c * HID + 2 * lane;
    atomicAdd(dst, v.x * coef);
    atomicAdd(dst + 1, v.y * coef);
  }
}

// ---------------------------- WMMA GEMM ------------------------------------
// C[M x 64] = A[M x K] * W[K x 64] (+ epilogue).
// Block: 256 threads = 8 waves; each wave owns a 16x64 tile (4 N-tiles of 16).
// K-loop steps of 4 using V_WMMA_F32_16X16X4_F32.
//
// A-fragment layout (ISA 7.12.2, 32-bit A 16x4):
//   lane 0-15:  v0 = A[m][kb+0], v1 = A[m][kb+1]
//   lane 16-31: v0 = A[m][kb+2], v1 = A[m][kb+3]      (m = lane & 15)
// B-fragment mirrors with K rows striped across the two half-waves.
// LDS holds W K-pair interleaved: W[k][col] at sW[(k>>1)*PAIR_STRIDE + col*2 + (k&1)],
// so a B fragment {W[k][col], W[k+1][col]} is one 8-byte ds_load_b64.
// C/D layout: vgpr r holds row r (lanes 0-15) / row r+8 (lanes 16-31), col = lane&15.
//
// MODE 0: C[idx]  = v + bias[col]
// MODE 1: C[idx]  = v;  C2[idx] = dinv[row]^2 * v + bias[col]   (t + agg-init)
// MODE 2: u = relu(aggin[idx] + v + bias[col]); C[idx] = u;
//         C2[idx] = first ? u : C2[idx] + u                      (h_new + x_local)

template <int K, int MODE>
__global__ void gemm_wmma_kernel(const float* __restrict__ A,
                                 const float* __restrict__ W,
                                 const float* __restrict__ bias,
                                 const float* __restrict__ dinv,
                                 const float* __restrict__ aggin,
                                 float* __restrict__ C,
                                 float* __restrict__ C2,
                                 int M, int firstLayer) {
  __shared__ float sW[(K / 2) * PAIR_STRIDE];

  // Stage W[K x 64] into LDS, K-pair interleaved.
  for (int i = threadIdx.x; i < K * HID; i += blockDim.x) {
    int k = i >> 6;
    int col = i & 63;
    sW[(k >> 1) * PAIR_STRIDE + col * 2 + (k & 1)] = W[i];
  }
  __syncthreads();

  int wave = threadIdx.x >> 5;
  int lane = threadIdx.x & 31;
  int mtile = blockIdx.x * 8 + wave;
  int row0 = mtile * 16;
  if (row0 >= M) return;  // wave-uniform: EXEC stays all-ones for WMMA

  int m = lane & 15;          // row within 16-row tile (A) / col within 16 (B, C)
  int kh = lane >> 4;         // which K-pair of the 4-wide K step this half-wave holds
  int kh2 = kh * 2;

  const float* Arow = A + ((size_t)row0 + m) * K;

  v8f acc0 = {}, acc1 = {}, acc2 = {}, acc3 = {};

#pragma unroll
  for (int kb = 0; kb < K; kb += 4) {
    v2f a = *(const v2f*)(Arow + kb + kh2);  // 8B load: A[m][kb+kh2 .. +1]
    const float* bp = &sW[(kb / 2 + kh) * PAIR_STRIDE + m * 2];
    v2f b0 = *(const v2f*)(bp + 0);    // cols m      : rows kb+kh2, kb+kh2+1
    v2f b1 = *(const v2f*)(bp + 32);   // cols m + 16
    v2f b2 = *(const v2f*)(bp + 64);   // cols m + 32
    v2f b3 = *(const v2f*)(bp + 96);   // cols m + 48
    acc0 = __builtin_amdgcn_wmma_f32_16x16x4_f32(false, a, false, b0, (short)0, acc0, false, false);
    acc1 = __builtin_amdgcn_wmma_f32_16x16x4_f32(false, a, false, b1, (short)0, acc1, false, false);
    acc2 = __builtin_amdgcn_wmma_f32_16x16x4_f32(false, a, false, b2, (short)0, acc2, false, false);
    acc3 = __builtin_amdgcn_wmma_f32_16x16x4_f32(false, a, false, b3, (short)0, acc3, false, false);
  }

#pragma unroll
  for (int r = 0; r < 8; ++r) {
    size_t row = (size_t)row0 + r + 8 * kh;
    size_t base = row * HID;
    float vals[4] = {acc0[r], acc1[r], acc2[r], acc3[r]};
    float d2 = 0.0f;
    if (MODE == 1) {
      float dv = dinv[row];
      d2 = dv * dv;
    }
#pragma unroll
    for (int nt = 0; nt < 4; ++nt) {
      int col = nt * 16 + m;
      size_t idx = base + col;
      float v = vals[nt];
      if (MODE == 0) {
        C[idx] = v + bias[col];
      } else if (MODE == 1) {
        C[idx] = v;                       // t = h @ W_gcn
        C2[idx] = d2 * v + bias[col];     // agg init: self-loop + b_gcn
      } else {                            // MODE == 2
        float u = aggin[idx] + v + bias[col];
        u = fmaxf(u, 0.0f);
        C[idx] = u;                       // h_new
        C2[idx] = firstLayer ? u : (C2[idx] + u);  // x_local accumulate
      }
    }
  }
}

// ---------------------------------------------------------------------------

extern "C" void kernel_launch(void* const* d_in, const int* in_sizes, int n_in,
                              void* d_out, int out_size, void* d_ws, size_t ws_size,
                              hipStream_t stream) {
  const float* x        = (const float*)d_in[0];
  const long long* ei   = (const long long*)d_in[1];  // int64 [2, E]
  const float* W_in     = (const float*)d_in[2];      // [128,64]
  const float* b_in     = (const float*)d_in[3];      // [64]
  const float* W_gcn    = (const float*)d_in[4];      // [3,64,64]
  const float* b_gcn    = (const float*)d_in[5];      // [3,64]
  const float* W_lin    = (const float*)d_in[6];      // [3,64,64]
  const float* b_lin    = (const float*)d_in[7];      // [3,64]
  const float* W_pred   = (const float*)d_in[8];      // [64,64]
  const float* b_pred   = (const float*)d_in[9];      // [64]
  float* out = (float*)d_out;

  const int N = N_NODES;
  const int E = N_EDGES;
  const long long* rows = ei;
  const long long* cols = ei + E;

  // Workspace carve-out: 5 * N*64 floats + N floats (~128.4 MB)
  size_t mat = (size_t)N * HID;
  float* hA   = (float*)d_ws;
  float* hB   = hA + mat;
  float* tbuf = hB + mat;
  float* agg  = tbuf + mat;
  float* xl   = agg + mat;
  float* dinv = xl + mat;

  // 1) symmetric-norm coefficients: dinv = rsqrt(deg_col + 1)
  zero_f32_kernel<<<(N + 255) / 256, 256, 0, stream>>>(dinv, N);
  deg_kernel<<<2048, 256, 0, stream>>>(cols, dinv, E);
  dinv_kernel<<<(N + 255) / 256, 256, 0, stream>>>(dinv, N);

  dim3 gemm_grid((N + 127) / 128);  // 8 waves * 16 rows per block

  // 2) h = x @ W_in + b_in
  gemm_wmma_kernel<128, 0><<<gemm_grid, 256, 0, stream>>>(
      x, W_in, b_in, nullptr, nullptr, hA, nullptr, N, 0);

  const float* hcur = hA;
  float* hnxt = hB;
  for (int i = 0; i < 3; ++i) {
    // 3a) t = h @ W_gcn[i];  agg = dinv^2 * t + b_gcn[i]   (self-loop + bias)
    gemm_wmma_kernel<64, 1><<<gemm_grid, 256, 0, stream>>>(
        hcur, W_gcn + (size_t)i * 4096, b_gcn + (size_t)i * 64,
        dinv, nullptr, tbuf, agg, N, 0);
    // 3b) agg[col] += t[row] * dinv[row]*dinv[col] over edges (L2-resident atomics)
    scatter_kernel<<<4096, 256, 0, stream>>>(rows, cols, dinv, tbuf, agg, E);
    // 3c) h = relu(agg + h @ W_lin[i] + b_lin[i]); x_local += h
    gemm_wmma_kernel<64, 2><<<gemm_grid, 256, 0, stream>>>(
        hcur, W_lin + (size_t)i * 4096, b_lin + (size_t)i * 64,
        nullptr, agg, hnxt, xl, N, (i == 0) ? 1 : 0);
    const float* tmp = hcur;
    hcur = hnxt;
    hnxt = (float*)tmp;
  }

  // 4) out = x_local @ W_pred + b_pred
  gemm_wmma_kernel<64, 0><<<gemm_grid, 256, 0, stream>>>(
      xl, W_pred, b_pred, nullptr, nullptr, out, nullptr, N, 0);
}